// OctVolSynth_26688926777587
// MI455X (gfx1250) — compile-verified
//
#include <hip/hip_runtime.h>
#include <hip/hip_bf16.h>

typedef int   i4 __attribute__((ext_vector_type(4)));
typedef float f4 __attribute__((ext_vector_type(4)));

#define LUT_MAX 512  // intensity_lut is 302 entries (labels 0..301); pad to 512

// Low 32 bits of a generic (flat) pointer into the LDS aperture are the LDS
// byte offset (ISA 10.2).
__device__ __forceinline__ unsigned lds_offset_of(const void* p) {
    return (unsigned)(unsigned long long)p;
}

// Vectorized main kernel: each thread handles 4 voxels via B128 NT loads/stores.
// The LUT is DMA'd straight into LDS with GLOBAL_LOAD_ASYNC_TO_LDS_B32 (no
// VGPR bounce, tracked by ASYNCcnt), then gathered per-lane with ds_load_b32.
__global__ __launch_bounds__(256) void octvolsynth_vec4(
    const int*   __restrict__ labels,
    const float* __restrict__ paren,
    const float* __restrict__ tex,
    const float* __restrict__ lut,
    int lut_n,
    float*       __restrict__ out_vol,
    float*       __restrict__ out_mask,
    long long n4)
{
    __shared__ float lut_s[LUT_MAX];

    // Async memory->LDS copy of the LUT (each active lane moves 4 bytes).
    for (int i = threadIdx.x; i < lut_n; i += blockDim.x) {
        unsigned     lds_off = lds_offset_of(&lut_s[i]);
        const float* gaddr   = lut + i;
        asm volatile("global_load_async_to_lds_b32 %0, %1, off"
                     :: "v"(lds_off), "v"(gaddr) : "memory");
    }
    // Drain this wave's async copies, then sync the workgroup.
    asm volatile("s_wait_asynccnt 0" ::: "memory");
    __syncthreads();

    long long idx = (long long)blockIdx.x * blockDim.x + threadIdx.x;
    if (idx >= n4) return;

    // Streaming, read-once inputs: non-temporal B128 loads.
    i4 lab = __builtin_nontemporal_load((const i4*)labels + idx);
    f4 p   = __builtin_nontemporal_load((const f4*)paren  + idx);
    f4 t   = __builtin_nontemporal_load((const f4*)tex    + idx);

    f4 vol, msk;
#pragma unroll
    for (int k = 0; k < 4; ++k) {
        float s = lut_s[lab[k]];          // ds_load_b32 gather
        float v = s * t[k];               // scaling * texture
        v = (v == 0.0f) ? 1.0f : v;       // vessels[vessels==0] = 1
        vol[k] = p[k] * v;                // parenchyma * vessels
        msk[k] = (lab[k] != 0) ? 1.0f : 0.0f;
    }

    // Streaming, write-once outputs: non-temporal B128 stores.
    __builtin_nontemporal_store(vol, (f4*)out_vol  + idx);
    __builtin_nontemporal_store(msk, (f4*)out_mask + idx);
}

// Scalar tail kernel for n % 4 != 0 (not hit for 256^3, kept for correctness).
__global__ __launch_bounds__(64) void octvolsynth_tail(
    const int*   __restrict__ labels,
    const float* __restrict__ paren,
    const float* __restrict__ tex,
    const float* __restrict__ lut,
    float*       __restrict__ out_vol,
    float*       __restrict__ out_mask,
    long long start, long long n)
{
    long long i = start + (long long)blockIdx.x * blockDim.x + threadIdx.x;
    if (i >= n) return;
    int   l = labels[i];
    float v = lut[l] * tex[i];
    v = (v == 0.0f) ? 1.0f : v;
    out_vol[i]  = paren[i] * v;
    out_mask[i] = (l != 0) ? 1.0f : 0.0f;
}

extern "C" void kernel_launch(void* const* d_in, const int* in_sizes, int n_in,
                              void* d_out, int out_size, void* d_ws, size_t ws_size,
                              hipStream_t stream) {
    (void)n_in; (void)d_ws; (void)ws_size; (void)out_size;

    const int*   labels = (const int*)  d_in[0];  // vessel_labels  int32 [256^3]
    const float* paren  = (const float*)d_in[1];  // parenchyma     f32   [256^3]
    const float* tex    = (const float*)d_in[2];  // vessel_texture f32   [256^3]
    const float* lut    = (const float*)d_in[3];  // intensity_lut  f32   [302]

    const long long n     = (long long)in_sizes[0];
    const int       lut_n = in_sizes[3];

    float* out_vol  = (float*)d_out;        // final_volume, n elements
    float* out_mask = (float*)d_out + n;    // vessel_mask (as 0.0/1.0), n elements

    const long long n4 = n >> 2;            // float4 groups
    if (n4 > 0) {
        const int block = 256;              // 8 wave32 waves per block
        const long long grid = (n4 + block - 1) / block;
        octvolsynth_vec4<<<(unsigned)grid, block, 0, stream>>>(
            labels, paren, tex, lut, lut_n, out_vol, out_mask, n4);
    }

    const long long done = n4 << 2;
    if (done < n) {
        const long long rem = n - done;
        const int block = 64;
        const long long grid = (rem + block - 1) / block;
        octvolsynth_tail<<<(unsigned)grid, block, 0, stream>>>(
            labels, paren, tex, lut, out_vol, out_mask, done, n);
    }
}